// LPCResidual_34205119545376
// MI455X (gfx1250) — compile-verified
//
#include <hip/hip_runtime.h>
#include <hip/hip_bf16.h>

// LPC residual: frame (N=320, hop=160) -> Hann window -> autocorr r[0..20]
// -> Levinson-Durbin (== Toeplitz solve with r0+EPS regularization)
// -> FIR b=[1,-a] -> IIR with a=[1]*21 -> clip[-1,1].
//
// Kernel 1: autocorrelation via V_WMMA_F32_16X16X4_F32, one frame per wave.
//   D[i,j] = sum_n xw[n+i]*xw[n-j] = r[i+j]  (zero-padded stream in both A and B)
// Kernel 2: Levinson + fused FIR/IIR, one frame per lane (32 frames/wave),
//   register pipeline for FIR, 20-phase register ring for IIR, LDS transpose
//   for coalesced output stores.

#define NFRM   320
#define HOP    160
#define SLEN   160000
#define FPB    1000            // frames per batch row
#define NB     64
#define TOTF   (NB * FPB)      // 64000 frames total
#define ORD    20
#define LPCP   (ORD + 1)
#define EPS_R  1e-5f

typedef float v2f __attribute__((ext_vector_type(2)));
typedef float v8f __attribute__((ext_vector_type(8)));

// ---------------------------------------------------------------------------
// Kernel 1: windowed autocorrelation via f32 WMMA. 8 frames per 256-thread
// block (one frame per wave32). Writes r[0..20] per frame to workspace.
// ---------------------------------------------------------------------------
__global__ __launch_bounds__(256) void k_autocorr_wmma(
    const float* __restrict__ x, float* __restrict__ r_ws) {
  __shared__ float win[NFRM];
  __shared__ float xw[8][384];   // padded stream: logical index n -> xw[w][32+n], pads zeroed
  __shared__ float rb[8][32];

  const int tid  = threadIdx.x;
  const int wave = tid >> 5;
  const int lane = tid & 31;

  // Hann window (jnp.hanning: 0.5 - 0.5*cos(2*pi*n/(N-1)))
  const float wc = 6.2831853071795864769f / 319.0f;
  for (int n = tid; n < NFRM; n += 256) win[n] = 0.5f - 0.5f * cosf(wc * (float)n);

  const int g    = blockIdx.x * 8 + wave;          // frame id, grid = 8000 blocks exactly
  const int b    = g / FPB;
  const int f    = g - b * FPB;
  const int base = b * SLEN + f * HOP;
  const int smax = SLEN - f * HOP;                 // samples n < smax are valid

  float* buf = xw[wave];
  buf[lane]       = 0.0f;                          // pad indices [-32..-1]
  buf[352 + lane] = 0.0f;                          // pad indices [320..351]
  __syncthreads();

  // stage windowed frame into LDS (coalesced reads)
  for (int c = 0; c < 10; ++c) {
    int n = c * 32 + lane;
    float v = (n < smax) ? x[base + n] : 0.0f;
    buf[32 + n] = v * win[n];
  }
  __syncthreads();

  // WMMA accumulation: A[i,t] = xw[4q+t+i], B[t,j] = xw[4q+t-j]
  // f32 16x16x4 layouts (ISA 7.12.2):
  //   A: lane l -> row m=l&15, a.x=K(2*hi), a.y=K(2*hi+1), hi=l>>4
  //   B: lane l -> col n=l&15, b.x=row 2*hi, b.y=row 2*hi+1
  const int m  = lane & 15;
  const int hi = lane >> 4;
  const int ab = 32 + 2 * hi + m;                  // A stream offset (rel. p=4q)
  const int bb = 32 + 2 * hi - m;                  // B stream offset

  v8f acc = {0.f, 0.f, 0.f, 0.f, 0.f, 0.f, 0.f, 0.f};
#pragma unroll 4
  for (int p = -16; p <= 332; p += 4) {            // 88 K-steps, covers full correlation
    v2f av, bv;
    av.x = buf[p + ab];  av.y = buf[p + ab + 1];
    bv.x = buf[p + bb];  bv.y = buf[p + bb + 1];
    acc = __builtin_amdgcn_wmma_f32_16x16x4_f32(
        /*neg_a=*/false, av, /*neg_b=*/false, bv,
        /*c_mod=*/(short)0, acc, /*reuse_a=*/false, /*reuse_b=*/false);
  }

  // D[lane, vgpr v] = r[(v + 8*hi) + m]. Deterministic unique writers:
  //   lanes 0..15 (hi==0): acc[0] = r[m]      -> rb[m]
  //   lane 15     (hi==0): acc[v] = r[15+v]   -> rb[16..20] for v=1..5
  if (hi == 0) {
    rb[wave][m] = acc[0];
    if (m == 15) {
      rb[wave][16] = acc[1]; rb[wave][17] = acc[2]; rb[wave][18] = acc[3];
      rb[wave][19] = acc[4]; rb[wave][20] = acc[5];
    }
  }
  __syncthreads();
  if (lane < LPCP) r_ws[g * LPCP + lane] = rb[wave][lane];
}

// ---------------------------------------------------------------------------
// Kernel 2: Levinson-Durbin + fused FIR/IIR, one frame per lane.
// grid = 250 blocks x 256 threads = 64000 frames exactly.
// ---------------------------------------------------------------------------
__global__ __launch_bounds__(256) void k_levinson_filter(
    const float* __restrict__ x, const float* __restrict__ r_ws,
    float* __restrict__ out) {
  __shared__ float ybuf[256][LPCP];   // 21-float stride: odd -> bank-conflict free

  const int tid  = threadIdx.x;
  const int g    = blockIdx.x * 256 + tid;
  const int b    = g / FPB;
  const int f    = g - b * FPB;
  const int base = b * SLEN + f * HOP;
  const int smax = SLEN - f * HOP;

  // ---- load autocorrelation, regularize diagonal ----
  float r[LPCP];
#pragma unroll
  for (int k = 0; k < LPCP; ++k) r[k] = r_ws[g * LPCP + k];
  r[0] += EPS_R;

  // ---- Levinson-Durbin: solve Toeplitz(r) a = r[1..20]  (fully unrolled) ----
  float a[LPCP];                       // a[1..20] used
  float E = r[0];
#pragma unroll
  for (int mm = 1; mm <= ORD; ++mm) {
    float s = r[mm];
#pragma unroll
    for (int j = 1; j < mm; ++j) s -= a[j] * r[mm - j];
    float kref = s / E;
    float tmp[LPCP];
#pragma unroll
    for (int j = 1; j < mm; ++j) tmp[j] = a[j] - kref * a[mm - j];
#pragma unroll
    for (int j = 1; j < mm; ++j) a[j] = tmp[j];
    a[mm] = kref;
    E *= (1.0f - kref * kref);
  }

  // ---- fused FIR (b = [1, -a]) + IIR (denominator all-ones) ----
  float gq[LPCP];                      // xf pipeline: gq[j] = xf[n-20+j]
#pragma unroll
  for (int j = 0; j < LPCP; ++j) gq[j] = 0.0f;
  gq[ORD] = x[base];                   // xf[0] (smax >= HOP > 0 always)

  float ring[ORD];                     // y[n-20..n-1], phase-indexed
#pragma unroll
  for (int j = 0; j < ORD; ++j) ring[j] = 0.0f;

  const int wave = tid >> 5;
  const int lane = tid & 31;

  for (int blk = 0; blk < 16; ++blk) {
#pragma unroll
    for (int ph = 0; ph < ORD; ++ph) {
      const int n = blk * ORD + ph;
      // FIR: fir[n] = xf[n] - sum_k a[k]*xf[n-k]
      float fir = gq[ORD];
#pragma unroll
      for (int k = 1; k <= ORD; ++k) fir -= a[k] * gq[ORD - k];
      // IIR: y[n] = fir[n] - sum(last 20 y)   (fresh sum, matches reference)
      float s = 0.0f;
#pragma unroll
      for (int k = 0; k < ORD; ++k) s += ring[k];
      float y = fir - s;
      ring[ph] = y;
      ybuf[tid][ph] = y;
      // advance pipeline
#pragma unroll
      for (int j = 0; j < ORD; ++j) gq[j] = gq[j + 1];
      const int nn = n + 1;
      float v = 0.0f;
      if (nn < NFRM && nn < smax) v = x[base + nn];
      gq[ORD] = v;
    }
    __syncthreads();
    // transposed, clipped, coalesced output: 20 consecutive samples per frame
    if (lane < ORD) {
      const int wb = wave * 32;
#pragma unroll 4
      for (int fi = 0; fi < 32; ++fi) {
        float y = ybuf[wb + fi][lane];
        y = fminf(1.0f, fmaxf(-1.0f, y));
        out[(blockIdx.x * 256 + wb + fi) * NFRM + blk * ORD + lane] = y;
      }
    }
    __syncthreads();
  }
}

// ---------------------------------------------------------------------------
extern "C" void kernel_launch(void* const* d_in, const int* in_sizes, int n_in,
                              void* d_out, int out_size, void* d_ws, size_t ws_size,
                              hipStream_t stream) {
  const float* x   = (const float*)d_in[0];
  float*       out = (float*)d_out;
  float*       rws = (float*)d_ws;      // 64000 * 21 floats = 5.4 MB scratch

  (void)in_sizes; (void)n_in; (void)out_size; (void)ws_size;

  k_autocorr_wmma   <<<TOTF / 8,   256, 0, stream>>>(x, rws);
  k_levinson_filter <<<TOTF / 256, 256, 0, stream>>>(x, rws, out);
}